// GCN_4793183502379
// MI455X (gfx1250) — compile-verified
//
#include <hip/hip_runtime.h>

#define N_NODES   100000
#define N_EDGES   1000000
#define DIM       64
#define N_GRAPHS  128
#define N_LAYERS  3
#define NTILES    (N_NODES / 16)   // 6250, exact

typedef __attribute__((ext_vector_type(2))) float v2f;
typedef __attribute__((ext_vector_type(8))) float v8f;

// --------------------------------------------------------------------------
// GEMM: Hlin[n, o] = sum_k H[n, k] * W[o, k]   (W row-major [64,64])
// One wave per 16-row tile; 4 column tiles of 16; K = 16 steps of 4 using
// V_WMMA_F32_16X16X4_F32 (fp32-exact vs reference).
// A 16x4 layout: lane<16 -> M=lane, K = {0,1}; lane>=16 -> M=lane-16, K={2,3}
// B 4x16 layout: lane<16 -> N=lane, K = {0,1}; lane>=16 -> N=lane-16, K={2,3}
// C/D 16x16:     vgpr i -> M = i + 8*(lane>=16), N = lane&15
// --------------------------------------------------------------------------
__global__ __launch_bounds__(256) void gcn_gemm_wmma(
    const float* __restrict__ H, const float* __restrict__ W,
    float* __restrict__ Hlin) {
  const int wave = blockIdx.x * (blockDim.x >> 5) + (threadIdx.x >> 5);
  if (wave >= NTILES) return;                     // uniform per wave: EXEC stays full
  const int lane = threadIdx.x & 31;
  const int half = lane >> 4;                     // 0 or 1
  const int r    = lane & 15;
  const int row0 = wave * 16;

  // Preload all 16 A fragments for this row tile (reused across 4 col tiles)
  v2f a[16];
  const float* hrow = H + (size_t)(row0 + r) * DIM + 2 * half;
#pragma unroll
  for (int kk = 0; kk < 16; ++kk)
    a[kk] = *(const v2f*)(hrow + kk * 4);         // K = 4*kk + 2*half + {0,1}

#pragma unroll
  for (int nt = 0; nt < 4; ++nt) {
    const int col0 = nt * 16;
    const float* wrow = W + (size_t)(col0 + r) * DIM + 2 * half;
    v8f c = {};
#pragma unroll
    for (int kk = 0; kk < 16; ++kk) {
      v2f b = *(const v2f*)(wrow + kk * 4);       // B[k][n] = W[n][k]
      c = __builtin_amdgcn_wmma_f32_16x16x4_f32(
              /*neg_a=*/false, a[kk], /*neg_b=*/false, b,
              /*c_mod=*/(short)0, c, /*reuse_a=*/false, /*reuse_b=*/false);
    }
#pragma unroll
    for (int i = 0; i < 8; ++i)
      Hlin[(size_t)(row0 + i + 8 * half) * DIM + col0 + r] = c[i];
  }
}

// --------------------------------------------------------------------------
// Edge scatter: one wave per edge, float2 per lane (64 features).
// Hagg[dst] += Hlin[src] * w   via hardware global_atomic_add_f32.
// --------------------------------------------------------------------------
__global__ __launch_bounds__(256) void gcn_scatter(
    const float* __restrict__ Hlin, const int* __restrict__ ei,
    const float* __restrict__ ea, float* __restrict__ Hagg) {
  const int e = blockIdx.x * 8 + (threadIdx.x >> 5);
  if (e >= N_EDGES) return;
  const int lane = threadIdx.x & 31;
  const int src = ei[e];
  const int dst = ei[N_EDGES + e];
  const float w = ea[e];
  v2f v = *(const v2f*)(Hlin + (size_t)src * DIM + lane * 2);
  float* p = Hagg + (size_t)dst * DIM + lane * 2;
  unsafeAtomicAdd(p + 0, v.x * w);
  unsafeAtomicAdd(p + 1, v.y * w);
}

// --------------------------------------------------------------------------
// Fused bias + PReLU + per-feature sum/sumsq reduction.
// Writes activated values back into H in place; LDS-reduces 4 per-feature
// partials per block, then 2 global atomics per feature per block.
// --------------------------------------------------------------------------
__global__ __launch_bounds__(256) void gcn_act_stats(
    float* __restrict__ H, const float* __restrict__ bias,
    const float* __restrict__ prelu_a, float* __restrict__ stats) {
  __shared__ float s_sum[256];
  __shared__ float s_sq[256];
  const int d = threadIdx.x & 63;
  const int g = threadIdx.x >> 6;                 // 0..3 row groups per block
  const float bv = bias[d];
  const float alpha = prelu_a[0];
  float sum = 0.f, sq = 0.f;
  for (int n = blockIdx.x * 4 + g; n < N_NODES; n += gridDim.x * 4) {
    float v = H[(size_t)n * DIM + d] + bv;
    v = (v >= 0.f) ? v : alpha * v;
    H[(size_t)n * DIM + d] = v;
    sum += v;
    sq += v * v;
  }
  s_sum[threadIdx.x] = sum;
  s_sq[threadIdx.x] = sq;
  __syncthreads();
  if (threadIdx.x < 64) {
    float ts = s_sum[threadIdx.x] + s_sum[64 + threadIdx.x] +
               s_sum[128 + threadIdx.x] + s_sum[192 + threadIdx.x];
    float tq = s_sq[threadIdx.x] + s_sq[64 + threadIdx.x] +
               s_sq[128 + threadIdx.x] + s_sq[192 + threadIdx.x];
    unsafeAtomicAdd(&stats[threadIdx.x], ts);
    unsafeAtomicAdd(&stats[64 + threadIdx.x], tq);
  }
}

// Fold mean/var/gamma/beta into per-feature scale (coef[d]) + shift (coef[64+d])
__global__ void gcn_bn_finalize(const float* __restrict__ stats,
                                const float* __restrict__ bn_w,
                                const float* __restrict__ bn_b,
                                float* __restrict__ coef) {
  const int d = threadIdx.x;
  if (d >= 64) return;
  const float inv_n = 1.0f / (float)N_NODES;
  float mean = stats[d] * inv_n;
  float var = stats[64 + d] * inv_n - mean * mean;
  float rstd = rsqrtf(var + 1e-5f);
  float A = rstd * bn_w[d];
  coef[d] = A;
  coef[64 + d] = bn_b[d] - mean * A;
}

// Normalize in place (H becomes next layer input) + graph pooling atomics
// into out[g*192 + layer*64 + d].
__global__ __launch_bounds__(256) void gcn_bn_apply_pool(
    float* __restrict__ H, const float* __restrict__ coef,
    const int* __restrict__ batch, float* __restrict__ out, int layer) {
  const size_t total = (size_t)N_NODES * DIM;
  for (size_t idx = (size_t)blockIdx.x * blockDim.x + threadIdx.x; idx < total;
       idx += (size_t)gridDim.x * blockDim.x) {
    const int d = (int)(idx & 63);
    const int n = (int)(idx >> 6);
    float v = H[idx] * coef[d] + coef[64 + d];
    H[idx] = v;
    const int gph = batch[n];
    unsafeAtomicAdd(&out[(size_t)gph * (N_LAYERS * DIM) + layer * DIM + d], v);
  }
}

// x_stack = pooled[last layer]: copy columns [128..191] of global_rep.
__global__ void gcn_copy_xstack(float* __restrict__ out) {
  const int idx = blockIdx.x * blockDim.x + threadIdx.x;
  if (idx < N_GRAPHS * DIM) {
    const int g = idx >> 6;
    const int d = idx & 63;
    out[N_GRAPHS * N_LAYERS * DIM + idx] =
        out[(size_t)g * (N_LAYERS * DIM) + (N_LAYERS - 1) * DIM + d];
  }
}

extern "C" void kernel_launch(void* const* d_in, const int* in_sizes, int n_in,
                              void* d_out, int out_size, void* d_ws, size_t ws_size,
                              hipStream_t stream) {
  const float* x        = (const float*)d_in[0];   // [N, 64]
  const int*   ei       = (const int*)d_in[1];     // [2, E]
  const float* ea       = (const float*)d_in[2];   // [E]
  const int*   batch    = (const int*)d_in[3];     // [N]
  const float* Wmat     = (const float*)d_in[4];   // [3, 64, 64]
  const float* bvec     = (const float*)d_in[5];   // [3, 64]
  const float* prelu_a  = (const float*)d_in[6];   // [1]
  const float* bn_w     = (const float*)d_in[7];   // [3, 64]
  const float* bn_b     = (const float*)d_in[8];   // [3, 64]
  float* out = (float*)d_out;                      // 128*192 + 128*64

  float* buf0  = (float*)d_ws;                     // lin output   [N, 64]
  float* buf1  = buf0 + (size_t)N_NODES * DIM;     // agg/act/next [N, 64]
  float* stats = buf1 + (size_t)N_NODES * DIM;     // 128 floats
  float* coef  = stats + 128;                      // 128 floats

  hipMemsetAsync(d_out, 0, sizeof(float) * (size_t)(N_GRAPHS * (N_LAYERS + 1) * DIM),
                 stream);

  const float* hin = x;
  for (int L = 0; L < N_LAYERS; ++L) {
    gcn_gemm_wmma<<<(NTILES + 7) / 8, 256, 0, stream>>>(hin, Wmat + L * DIM * DIM, buf0);
    hipMemsetAsync(buf1, 0, sizeof(float) * (size_t)N_NODES * DIM, stream);
    hipMemsetAsync(stats, 0, sizeof(float) * 128, stream);
    gcn_scatter<<<N_EDGES / 8, 256, 0, stream>>>(buf0, ei, ea, buf1);
    gcn_act_stats<<<512, 256, 0, stream>>>(buf1, bvec + L * DIM, prelu_a, stats);
    gcn_bn_finalize<<<1, 64, 0, stream>>>(stats, bn_w + L * DIM, bn_b + L * DIM, coef);
    gcn_bn_apply_pool<<<2048, 256, 0, stream>>>(buf1, coef, batch, out, L);
    hin = buf1;                                    // normalized output feeds next layer
  }
  gcn_copy_xstack<<<(N_GRAPHS * DIM + 255) / 256, 256, 0, stream>>>(out);
}